// DeformableConvStandard_84207128805830
// MI455X (gfx1250) — compile-verified
//
#include <hip/hip_runtime.h>
#include <hip/hip_bf16.h>

typedef __attribute__((ext_vector_type(2))) float v2f;
typedef __attribute__((ext_vector_type(8))) float v8f;

#define N_PRED   12
#define N_DW     3
#define LWIN     16            // L = N_PRED + 2*N_DRIFT
#define KX       48            // N_DW * LWIN, the contraction dim
#define N_NODES  4096
#define BATCH    128
#define NROWS    (BATCH * N_NODES)   // 524288

// d_ws layout (floats):
//   [0    .. 767 ]  Bn fragments : 12 k-steps * 64 (lane*2 + v)
//   [768  .. 1535]  Bt fragments : 12 k-steps * 64
//   [1536 .. 1727]  Wg fragments :  3 k-steps * 64
#define WS_BN 0
#define WS_BT 768
#define WS_WG 1536

// ---------------------------------------------------------------------------
// Stage 0: fold the constant deform (tanh/floor/lerp of offsets) and the conv
// weights into two dense 48x16 matrices (cols 12..15 zero-padded), plus the
// zero-padded gate matrix W, stored directly in WMMA B-fragment order.
// Fragment order for V_WMMA_F32_16X16X4_F32 B (4x16, 2 VGPRs):
//   lane<16 : v0=B[4j+0, lane], v1=B[4j+1, lane]
//   lane>=16: v0=B[4j+2, lane-16], v1=B[4j+3, lane-16]
// ---------------------------------------------------------------------------
__global__ void build_frags_kernel(const float* __restrict__ offset_t,
                                   const float* __restrict__ offset_n,
                                   const float* __restrict__ conv_t_w,
                                   const float* __restrict__ conv_n_w,
                                   const float* __restrict__ W,
                                   float* __restrict__ ws)
{
    if (threadIdx.x != 0 || blockIdx.x != 0) return;

    float Mn[KX][16];
    float Mt[KX][16];
    for (int r = 0; r < KX; ++r)
        for (int c = 0; c < 16; ++c) { Mn[r][c] = 0.0f; Mt[r][c] = 0.0f; }

    for (int k = 0; k < N_DW; ++k) {
        for (int c = 0; c < N_PRED; ++c) {
            // ----- offset_n path: pred_n[o] += dn[k][c] * conv_n_w[o][c][k]
            {
                float pos = tanhf(offset_n[k * N_PRED + c]) * 2.0f + (float)(c + 2);
                float kf  = floorf(pos);
                int   idx = (int)kf;
                float fr  = pos - kf;
                for (int o = 0; o < N_PRED; ++o) {
                    float w = conv_n_w[o * (N_PRED * N_DW) + c * N_DW + k];
                    Mn[k * LWIN + idx    ][o] += (1.0f - fr) * w;
                    Mn[k * LWIN + idx + 1][o] += fr * w;
                }
            }
            // ----- offset_t path: cross-correlation, pad 1:
            // pred_t[p] = sum_{k,d} conv_t_w[k][d] * dt[k][p + d - 1]
            // so dt[k][c] feeds p = c + 1 - d for d in 0..2 (p in [0,12))
            {
                float pos = tanhf(offset_t[k * N_PRED + c]) * 2.0f + (float)(c + 2);
                float kf  = floorf(pos);
                int   idx = (int)kf;
                float fr  = pos - kf;
                for (int d = 0; d < 3; ++d) {
                    int p = c + 1 - d;
                    if (p < 0 || p >= N_PRED) continue;
                    float w = conv_t_w[k * 3 + d];
                    Mt[k * LWIN + idx    ][p] += (1.0f - fr) * w;
                    Mt[k * LWIN + idx + 1][p] += fr * w;
                }
            }
        }
    }

    // Emit Bn / Bt fragments
    for (int j = 0; j < 12; ++j) {
        for (int ln = 0; ln < 32; ++ln) {
            int row = 4 * j + ((ln >= 16) ? 2 : 0);
            int col = ln & 15;
            ws[WS_BN + j * 64 + ln * 2 + 0] = Mn[row][col];
            ws[WS_BN + j * 64 + ln * 2 + 1] = Mn[row + 1][col];
            ws[WS_BT + j * 64 + ln * 2 + 0] = Mt[row][col];
            ws[WS_BT + j * 64 + ln * 2 + 1] = Mt[row + 1][col];
        }
    }
    // Emit gate W fragments (K = 12 -> 3 k-steps, cols 12..15 zero)
    for (int j = 0; j < 3; ++j) {
        for (int ln = 0; ln < 32; ++ln) {
            int row = 4 * j + ((ln >= 16) ? 2 : 0);
            int col = ln & 15;
            float w0 = (col < N_PRED) ? W[row * N_PRED + col]       : 0.0f;
            float w1 = (col < N_PRED) ? W[(row + 1) * N_PRED + col] : 0.0f;
            ws[WS_WG + j * 64 + ln * 2 + 0] = w0;
            ws[WS_WG + j * 64 + ln * 2 + 1] = w1;
        }
    }
}

// ---------------------------------------------------------------------------
// Stage 1: fused GEMM + gate + blend. One wave = 32 rows (two 16-row tiles).
// 54 v_wmma_f32_16x16x4_f32 per wave; epilogue is ~30 VALU flops/row.
// ---------------------------------------------------------------------------
__global__ __launch_bounds__(256) void fused_main_kernel(
    const float* __restrict__ inp,
    const float* __restrict__ ctrl,
    const float* __restrict__ conv_t_b,
    const float* __restrict__ conv_n_b,
    const float* __restrict__ bparam,
    const float* __restrict__ frags,
    float* __restrict__ out)
{
    const int lane  = threadIdx.x & 31;
    const int wave  = threadIdx.x >> 5;
    const int wpb   = blockDim.x >> 5;                    // 8 waves/block
    const int r0    = (blockIdx.x * wpb + wave) * 32;     // first of 32 rows

    const int mrow  = lane & 15;          // M index within tile
    const int khalf = (lane >> 4) & 1;    // 0: K pair {0,1}, 1: K pair {2,3}

    // Constant B fragments (hot in L2 / WGP$ across all waves)
    v2f Bn[12], Bt[12], Wg[3];
#pragma unroll
    for (int j = 0; j < 12; ++j) {
        Bn[j] = *(const v2f*)(frags + WS_BN + j * 64 + lane * 2);
        Bt[j] = *(const v2f*)(frags + WS_BT + j * 64 + lane * 2);
    }
#pragma unroll
    for (int j = 0; j < 3; ++j)
        Wg[j] = *(const v2f*)(frags + WS_WG + j * 64 + lane * 2);

    v8f accN[2] = {v8f{}, v8f{}};
    v8f accT[2] = {v8f{}, v8f{}};
    v8f accG[2] = {v8f{}, v8f{}};

#pragma unroll
    for (int t = 0; t < 2; ++t) {
        // A fragments of x: row-major 48 floats per row
        const float* xbase = inp + (size_t)(r0 + t * 16 + mrow) * KX + khalf * 2;
#pragma unroll
        for (int j = 0; j < 12; ++j) {
            v2f a = *(const v2f*)(xbase + j * 4);
            accN[t] = __builtin_amdgcn_wmma_f32_16x16x4_f32(
                false, a, false, Bn[j], (short)0, accN[t], false, false);
            accT[t] = __builtin_amdgcn_wmma_f32_16x16x4_f32(
                false, a, false, Bt[j], (short)0, accT[t], false, false);
        }
        // A fragments of ctrl: 12 floats per row
        const float* cbase = ctrl + (size_t)(r0 + t * 16 + mrow) * N_PRED + khalf * 2;
#pragma unroll
        for (int j = 0; j < 3; ++j) {
            v2f a = *(const v2f*)(cbase + j * 4);
            accG[t] = __builtin_amdgcn_wmma_f32_16x16x4_f32(
                false, a, false, Wg[j], (short)0, accG[t], false, false);
        }
    }

    // Epilogue. C/D layout: VGPR r, lanes 0-15 -> (M=r, N=lane),
    // lanes 16-31 -> (M=r+8, N=lane-16).
    const float tb  = conv_t_b[0];
    const int   col = lane & 15;
    const int   mhi = (lane >= 16) ? 8 : 0;
    if (col < N_PRED) {
        const float nb = conv_n_b[col];
#pragma unroll
        for (int t = 0; t < 2; ++t) {
#pragma unroll
            for (int r = 0; r < 8; ++r) {
                int   row  = r0 + t * 16 + r + mhi;
                int   node = row & (N_NODES - 1);
                float g    = accG[t][r] + bparam[node * N_PRED + col];
                g          = 1.0f / (1.0f + __expf(-g));
                float pn   = accN[t][r] + nb;
                float pt   = accT[t][r] + tb;
                out[(size_t)row * N_PRED + col] = pn * g + pt * (1.0f - g);
            }
        }
    }
}

extern "C" void kernel_launch(void* const* d_in, const int* in_sizes, int n_in,
                              void* d_out, int out_size, void* d_ws, size_t ws_size,
                              hipStream_t stream)
{
    // setup_inputs() order:
    // 0 inp, 1 ctrl, 2 offset_t, 3 offset_n, 4 conv_t_w, 5 conv_t_b,
    // 6 conv_n_w, 7 conv_n_b, 8 W, 9 bparam
    const float* inp      = (const float*)d_in[0];
    const float* ctrl     = (const float*)d_in[1];
    const float* offset_t = (const float*)d_in[2];
    const float* offset_n = (const float*)d_in[3];
    const float* conv_t_w = (const float*)d_in[4];
    const float* conv_t_b = (const float*)d_in[5];
    const float* conv_n_w = (const float*)d_in[6];
    const float* conv_n_b = (const float*)d_in[7];
    const float* W        = (const float*)d_in[8];
    const float* bparam   = (const float*)d_in[9];
    float*       outp     = (float*)d_out;
    float*       ws       = (float*)d_ws;

    build_frags_kernel<<<1, 32, 0, stream>>>(offset_t, offset_n, conv_t_w,
                                             conv_n_w, W, ws);

    // 32 rows per wave, 8 waves per block -> 256 rows per block
    const int blocks = NROWS / 256;   // 2048
    fused_main_kernel<<<blocks, 256, 0, stream>>>(inp, ctrl, conv_t_b, conv_n_b,
                                                  bparam, ws, outp);
}